// FBPinn_58196806861032
// MI455X (gfx1250) — compile-verified
//
#include <hip/hip_runtime.h>
#include <hip/hip_bf16.h>

// FBPiNN forward, fp32, WMMA f32 16x16x4 on gfx1250 (wave32).
// N=65536 pts, 16 windows, layers 1->128, 128->128 x2, 128->1.
// Wave-per-column-tile GEMM: B fragments register-resident per layer,
// dual-accumulator WMMA chains, opaque LDS row bases (offset-folded ds),
// double-buffered async (global->LDS) weight staging on ASYNCcnt.

#define NW    16
#define NEUR  128
#define NHID  2
#define NPTS  65536
#define SIGMA 0.02f

#define TILE_N  128
#define THREADS 256   // 8 wave32 waves
#define HPAD    4     // Hs row stride 132 (132 % 64 == 4): conflict-free A-frag loads
#define HROW    (NEUR + HPAD)

typedef float v2f __attribute__((ext_vector_type(2)));
typedef float v8f __attribute__((ext_vector_type(8)));
typedef int   v4i __attribute__((ext_vector_type(4)));

#if __has_builtin(__builtin_amdgcn_global_load_async_to_lds_b128) && \
    __has_builtin(__builtin_amdgcn_s_wait_asynccnt)
#define USE_ASYNC 1
#else
#define USE_ASYNC 0
#endif

typedef v4i __attribute__((address_space(1)))* gb128_p;   // global int4*
typedef v4i __attribute__((address_space(3)))* lb128_p;   // LDS int4*
typedef float __attribute__((address_space(3)))* lfp;     // LDS float*
typedef v2f   __attribute__((address_space(3)))* lv2f_p;  // LDS float2*

struct Smem {
    float Ws[2][NEUR][NEUR];          // at LDS offset 0: B-frag ds offsets < 64KB
    float Hs[2][TILE_N][HROW];        // ping-pong activations (132-float rows)
    float xs[TILE_N];
};

// Stage one 64KB weight tile into LDS. Async path: 16 x b128 per thread,
// contiguous copy, tracked on ASYNCcnt; sync fallback: float4 copy.
__device__ __forceinline__ void stage_weights(const float* __restrict__ Wg,
                                              float* lds, int tid) {
#if USE_ASYNC
    const char* g = (const char*)Wg;
    char*       l = (char*)lds;
    #pragma unroll
    for (int i = 0; i < (NEUR * NEUR * 4) / (THREADS * 16); ++i) {   // 16 iters
        const int off = (tid + i * THREADS) * 16;
        __builtin_amdgcn_global_load_async_to_lds_b128(
            (gb128_p)(g + off),
            (lb128_p)(l + off),
            0, 0);
    }
#else
    const float4* g4 = (const float4*)Wg;
    float4*       l4 = (float4*)lds;
    #pragma unroll 4
    for (int i = tid; i < (NEUR * NEUR) / 4; i += THREADS) l4[i] = g4[i];
#endif
}

__device__ __forceinline__ void wait_stage() {
#if USE_ASYNC
    // 16 async b128 just issued for the NEXT tile may remain outstanding;
    // wait for the previous 16 (this wave's share of the CURRENT tile).
    __builtin_amdgcn_s_wait_asynccnt(16);
#endif
}

__global__ __launch_bounds__(THREADS)
void fbpinn_fwd_kernel(const float* __restrict__ x,
                       const float* __restrict__ means,
                       const float* __restrict__ stds,
                       const float* __restrict__ mids,
                       const float* __restrict__ W_in,
                       const float* __restrict__ b_in,
                       const float* __restrict__ W_hid,
                       const float* __restrict__ b_hid,
                       const float* __restrict__ W_out,
                       const float* __restrict__ b_out,
                       float* __restrict__ out)
{
    extern __shared__ char smem_raw[];
    Smem& sm = *reinterpret_cast<Smem*>(smem_raw);

    const int tid  = threadIdx.x;
    const int lane = tid & 31;
    const int wv   = tid >> 5;          // wave id 0..7 == owned column tile
    const int n0   = blockIdx.x * TILE_N;

    const int hi   = lane >> 4;         // half-wave select (0/1)
    const int lo   = lane & 15;
    const int eB   = (wv << 4) + lo;    // this lane's B-fragment / output column

    // stage x tile
    if (tid < TILE_N) sm.xs[tid] = x[n0 + tid];

    // prime the pipeline: weights for (w=0, l=0) -> buffer 0
    stage_weights(W_hid, &sm.Ws[0][0][0], tid);
    int buf = 0;

    // even lanes own one point each for output accumulation
    const int p_own = (wv << 4) + (lane >> 1);
    float racc = 0.0f;

    for (int w = 0; w < NW; ++w) {
        __syncthreads();  // previous window's readers of Hs[0] / xs done

        // ---------------- input layer (elementwise) -> Hs[0] -------------
        const float mean_w = means[w];
        const float istd_w = 1.0f / stds[w];
        {
            const int e  = tid & (NEUR - 1);
            const int pg = tid >> 7;                 // 0 or 1
            const float wi = W_in[w * NEUR + e];
            const float bi = b_in[w * NEUR + e];
            #pragma unroll 4
            for (int i = 0; i < TILE_N / 2; ++i) {
                const int p  = pg * (TILE_N / 2) + i;
                const float xn = (sm.xs[p] - mean_w) * istd_w;
                sm.Hs[0][p][e] = tanhf(xn * wi + bi);
            }
        }

        // ---------------- hidden layers: Hdst = tanh(Hsrc*W + b) --------
        for (int l = 0; l < NHID; ++l) {
            __syncthreads();  // Hs writes visible; Ws[buf^1] readers done

            // async prefetch of the NEXT step's weights (wraps at end)
            {
                const int wn = (l == 0) ? w : ((w + 1) & (NW - 1));
                const int ln = (l == 0) ? 1 : 0;
                const float* Wgn = W_hid + (size_t)(ln * NW + wn) * (NEUR * NEUR);
                stage_weights(Wgn, &sm.Ws[buf ^ 1][0][0], tid);
            }
            wait_stage();     // this wave's share of the CURRENT tile has landed
            __syncthreads();  // every wave's share has landed

            // B fragments for the owned column tile: register-resident all layer.
            // fp32 4x16 B layout: lanes0-15 rows (4k,4k+1), lanes16-31 (4k+2,4k+3)
            lfp bcol = (lfp)(&sm.Ws[buf][0][0] + eB);
            asm("" : "+v"(bcol));            // opaque base -> immediate ds offsets
            v2f b[32];
            #pragma unroll
            for (int k = 0; k < 32; ++k) {
                const int d0 = 4 * k + 2 * hi;
                b[k].x = bcol[d0 * NEUR];
                b[k].y = bcol[d0 * NEUR + NEUR];
            }
            const float bias = b_hid[(l * NW + w) * NEUR + eB];

            const float* Hsrc = &sm.Hs[l & 1][0][0];
            float*       Hdst = &sm.Hs[(l & 1) ^ 1][0][0];

            for (int r = 0; r < 8; ++r) {
                // A fragments for row tile r (adjacent K pairs -> b64 loads)
                lfp arow = (lfp)(Hsrc + (16 * r + lo) * HROW + 2 * hi);
                asm("" : "+v"(arow));        // opaque base -> offsets 16k, fusable
                v2f a[32];
                #pragma unroll
                for (int k = 0; k < 32; ++k)
                    a[k] = *(lv2f_p)(arow + 4 * k);

                // dual accumulators: 2x ILP on the WMMA dependency chain
                v8f acc0 = {0.f, 0.f, 0.f, 0.f, 0.f, 0.f, 0.f, 0.f};
                v8f acc1 = {0.f, 0.f, 0.f, 0.f, 0.f, 0.f, 0.f, 0.f};
                #pragma unroll
                for (int k = 0; k < 32; k += 2) {
                    acc0 = __builtin_amdgcn_wmma_f32_16x16x4_f32(
                        false, a[k],     false, b[k],     (short)0, acc0, false, false);
                    acc1 = __builtin_amdgcn_wmma_f32_16x16x4_f32(
                        false, a[k + 1], false, b[k + 1], (short)0, acc1, false, false);
                }
                v8f accs = acc0 + acc1;

                // D layout: VGPR v -> point 16r + v + 8*hi, column eB
                lfp drow = (lfp)(Hdst + (16 * r + 8 * hi) * HROW + eB);
                asm("" : "+v"(drow));        // opaque base -> immediate ds offsets
                #pragma unroll
                for (int v = 0; v < 8; ++v)
                    drow[v * HROW] = tanhf(accs[v] + bias);
            }
            buf ^= 1;
        }

        __syncthreads();  // Hs[0] (written by all waves in layer l=1) coherent

        // ---------------- output layer + window --------------------------
        {
            const int   p    = (wv << 4) + (lane >> 1);
            const int   e0   = (lane & 1) << 6;        // two lanes split the 128-dot
            const float* hrow = &sm.Hs[0][p][e0];
            const float* wo   = W_out + w * NEUR + e0;
            float s = 0.0f;
            #pragma unroll
            for (int j = 0; j < 64; j += 4) {
                float4 hv  = *reinterpret_cast<const float4*>(hrow + j);
                float4 wv4 = *reinterpret_cast<const float4*>(wo + j);
                s += hv.x * wv4.x + hv.y * wv4.y + hv.z * wv4.z + hv.w * wv4.w;
            }
            s += __shfl_xor(s, 1, 32);                 // combine half-sums
            if ((lane & 1) == 0) {
                const float u  = s + b_out[w];         // U_SD=1, U_MEAN=0
                const float xv = sm.xs[p];
                const float tl = (xv - mids[w])     * (1.0f / SIGMA);
                const float tr = (xv - mids[w + 1]) * (1.0f / SIGMA);
                const float wnd = (1.0f / (1.0f + __expf(tl))) *
                                  (1.0f / (1.0f + __expf(-tr)));
                racc += wnd * u;
            }
        }
    }

    if ((lane & 1) == 0) out[n0 + p_own] = racc;
}

extern "C" void kernel_launch(void* const* d_in, const int* in_sizes, int n_in,
                              void* d_out, int out_size, void* d_ws, size_t ws_size,
                              hipStream_t stream) {
    const float* x     = (const float*)d_in[0];
    const float* means = (const float*)d_in[1];
    const float* stds  = (const float*)d_in[2];
    const float* mids  = (const float*)d_in[3];
    const float* W_in  = (const float*)d_in[4];
    const float* b_in  = (const float*)d_in[5];
    const float* W_hid = (const float*)d_in[6];
    const float* b_hid = (const float*)d_in[7];
    const float* W_out = (const float*)d_in[8];
    const float* b_out = (const float*)d_in[9];
    float* outp = (float*)d_out;

    dim3 grid(NPTS / TILE_N);
    dim3 block(THREADS);
    size_t shmem = sizeof(Smem);   // ~260.5 KB of the 320 KB/WGP LDS
    fbpinn_fwd_kernel<<<grid, block, shmem, stream>>>(
        x, means, stds, mids, W_in, b_in, W_hid, b_hid, W_out, b_out, outp);
}